// ConAM_65472481460695
// MI455X (gfx1250) — compile-verified
//
#include <hip/hip_runtime.h>
#include <cmath>

typedef __attribute__((ext_vector_type(2))) float v2f;
typedef __attribute__((ext_vector_type(8))) float v8f;

#define Cc   256
#define HH   256
#define WW   256
#define BB   8
#define NPAT 256              // 16x16 patches
#define NROW 257              // patches + global token
#define MROWS (BB * NROW)     // 2056 real GEMM rows
#define MPAD  2064            // padded to 129 * 16 tiles (pad rows are dead data)

// ---------------------------------------------------------------------------
// Kernel 1: patch means (16x16) + plane mean for one (b,c) plane per block.
// 256 threads: thread t owns float4 column (t&63) over rows [(t>>6)*64, +64).
// Fully coalesced float4 streaming of the 536MB input.
// Writes mix[(b*257 + n)*256 + c] (n=0..255 patches, n=256 global).
// ---------------------------------------------------------------------------
__global__ void pool_kernel(const float* __restrict__ x, float* __restrict__ mix) {
    const int bc = blockIdx.x;            // b*256 + c
    const int b  = bc >> 8;
    const int c  = bc & 255;
    const float4* plane = reinterpret_cast<const float4*>(x + (size_t)bc * (HH * WW));

    const int t    = threadIdx.x;
    const int col4 = t & 63;              // float4 column within row
    const int rg   = t >> 6;              // row group 0..3 (rows rg*64..rg*64+63)

    float4 csum[4];
#pragma unroll
    for (int i = 0; i < 4; ++i) { csum[i].x = csum[i].y = csum[i].z = csum[i].w = 0.f; }

#pragma unroll 4
    for (int r = 0; r < 64; ++r) {
        const int row = rg * 64 + r;
        float4 v = plane[row * 64 + col4];
        const int pl = r >> 4;            // local patch-row 0..3
        csum[pl].x += v.x; csum[pl].y += v.y; csum[pl].z += v.z; csum[pl].w += v.w;
    }

    __shared__ float4 lds[16][64];        // [patch-row][float4 column] = 16KB
#pragma unroll
    for (int i = 0; i < 4; ++i) lds[rg * 4 + i][col4] = csum[i];
    __syncthreads();

    // thread t computes patch n = t  (ph = t>>4, pw = t&15)
    const int ph = t >> 4, pw = t & 15;
    float4 s4; s4.x = s4.y = s4.z = s4.w = 0.f;
#pragma unroll
    for (int j = 0; j < 4; ++j) {
        float4 v = lds[ph][pw * 4 + j];
        s4.x += v.x; s4.y += v.y; s4.z += v.z; s4.w += v.w;
    }
    const float psum = s4.x + s4.y + s4.z + s4.w;

    mix[((size_t)b * NROW + t) * Cc + c] = psum * (1.0f / 256.0f);

    // block-reduce psum -> plane sum for the global token
    __shared__ float red[256];
    red[t] = psum;
    __syncthreads();
    for (int off = 128; off > 0; off >>= 1) {
        if (t < off) red[t] += red[t + off];
        __syncthreads();
    }
    if (t == 0)
        mix[((size_t)b * NROW + NPAT) * Cc + c] = red[0] * (1.0f / 65536.0f);
}

// ---------------------------------------------------------------------------
// Kernel 2/3: out = relu(in @ w^T + bias), in: (MPAD,256), w: (256,256) o-major.
// One wave per 16x16 output tile, V_WMMA_F32_16X16X4_F32, K-loop of 64 steps.
// M padded to a tile multiple so the inner loop is branch-free:
//   2x global_load_b64 + 1x v_wmma per K step, no exec-mask manipulation.
// Pad-row garbage only reaches pad output rows, which are never consumed.
// A frag: lane l -> row m0+(l&15), K = k+2*(l>>4)..+1  (float2 load)
// B frag: lane l -> col n0+(l&15), K = k+2*(l>>4)..+1  (float2 from w[o][k])
// D: vgpr g -> row m0+g+8*(l>>4), col n0+(l&15)
// ---------------------------------------------------------------------------
__global__ void gemm_relu_kernel(const float* __restrict__ in,
                                 const float* __restrict__ w,
                                 const float* __restrict__ bias,
                                 float* __restrict__ out) {
    const int waves_per_block = blockDim.x >> 5;
    const int tile = blockIdx.x * waves_per_block + (threadIdx.x >> 5);
    if (tile >= (MPAD / 16) * 16) return; // wave-uniform exit: EXEC full for WMMA

    const int m0 = (tile >> 4) << 4;      // 16 N-tiles per M-tile
    const int n0 = (tile & 15) << 4;

    const int lane = threadIdx.x & 31;
    const int half = lane >> 4;           // 0|1
    const int l16  = lane & 15;

    const float* aptr = in + (size_t)(m0 + l16) * Cc + 2 * half;
    const float* bptr = w  + (size_t)(n0 + l16) * Cc + 2 * half;

    v8f acc = {};
#pragma unroll 16
    for (int k = 0; k < Cc; k += 4) {
        v2f a  = *reinterpret_cast<const v2f*>(aptr + k);
        v2f bv = *reinterpret_cast<const v2f*>(bptr + k);
        acc = __builtin_amdgcn_wmma_f32_16x16x4_f32(
            /*neg_a=*/false, a, /*neg_b=*/false, bv,
            /*c_mod=*/(short)0, acc, /*reuse_a=*/false, /*reuse_b=*/false);
    }

    const float bn = bias[n0 + l16];
#pragma unroll
    for (int g = 0; g < 8; ++g) {
        const int row = m0 + g + 8 * half;
        float v = acc[g] + bn;
        out[(size_t)row * Cc + (n0 + l16)] = v > 0.f ? v : 0.f;
    }
}

// ---------------------------------------------------------------------------
// Kernel 4: per batch: scores[n] = <m2[b,n,:], m2[b,256,:]>, softmax over n.
// One block per b, 256 threads (thread n owns one patch row).
// ---------------------------------------------------------------------------
__global__ void score_softmax_kernel(const float* __restrict__ m2,
                                     float* __restrict__ att) {
    const int b = blockIdx.x;
    const int n = threadIdx.x;

    __shared__ float gf[Cc];
    __shared__ float red[256];

    gf[n] = m2[((size_t)b * NROW + NPAT) * Cc + n];
    __syncthreads();

    const float* row = m2 + ((size_t)b * NROW + n) * Cc;
    float s = 0.f;
#pragma unroll 8
    for (int c = 0; c < Cc; ++c) s += row[c] * gf[c];

    // max-reduce
    red[n] = s;
    __syncthreads();
    for (int off = 128; off > 0; off >>= 1) {
        if (n < off) red[n] = fmaxf(red[n], red[n + off]);
        __syncthreads();
    }
    const float mx = red[0];
    __syncthreads();

    const float e = __expf(s - mx);
    red[n] = e;
    __syncthreads();
    for (int off = 128; off > 0; off >>= 1) {
        if (n < off) red[n] += red[n + off];
        __syncthreads();
    }
    att[b * NPAT + n] = e / red[0];
}

// ---------------------------------------------------------------------------
// Kernel 5: out = x * (1 + att[b, (h/16)*16 + w/16]), float4 streaming.
// ---------------------------------------------------------------------------
__global__ void apply_att_kernel(const float* __restrict__ x,
                                 const float* __restrict__ att,
                                 float* __restrict__ out) {
    const int i = blockIdx.x * blockDim.x + threadIdx.x;   // float4 index
    const int e = i << 2;                                  // element index (<2^27)
    const int w = e & 255;
    const int h = (e >> 8) & 255;
    const int b = e >> 24;

    const float a = 1.0f + att[(b << 8) + ((h >> 4) << 4) + (w >> 4)];
    float4 v = reinterpret_cast<const float4*>(x)[i];
    v.x *= a; v.y *= a; v.z *= a; v.w *= a;
    reinterpret_cast<float4*>(out)[i] = v;
}

// ---------------------------------------------------------------------------
extern "C" void kernel_launch(void* const* d_in, const int* in_sizes, int n_in,
                              void* d_out, int out_size, void* d_ws, size_t ws_size,
                              hipStream_t stream) {
    const float* x  = (const float*)d_in[0];
    const float* w1 = (const float*)d_in[1];
    const float* b1 = (const float*)d_in[2];
    const float* w2 = (const float*)d_in[3];
    const float* b2 = (const float*)d_in[4];
    float* out = (float*)d_out;

    const size_t buf_elems = (size_t)MPAD * Cc;            // 528384 floats each
    float* mix = (float*)d_ws;
    float* m1  = mix + buf_elems;
    float* m2  = m1 + buf_elems;
    float* att = m2 + buf_elems;                           // 8*256 floats

    // 1) pooling: one block per (b,c) plane
    pool_kernel<<<BB * Cc, 256, 0, stream>>>(x, mix);

    // 2,3) MLP: 129 Mtiles * 16 Ntiles = 2064 tiles; 8 waves/block -> 258 blocks
    const int tiles  = (MPAD / 16) * 16;
    const int blocks = (tiles + 7) / 8;
    gemm_relu_kernel<<<blocks, 256, 0, stream>>>(mix, w1, b1, m1);
    gemm_relu_kernel<<<blocks, 256, 0, stream>>>(m1, w2, b2, m2);

    // 4) attention scores + softmax: one block per batch
    score_softmax_kernel<<<BB, 256, 0, stream>>>(m2, att);

    // 5) apply: 2^27 elements / 4 per thread
    const int n4 = (BB * Cc * HH * WW) / 4;                // 33554432
    apply_att_kernel<<<n4 / 256, 256, 0, stream>>>(x, att, out);
}